// ODEBlock_51462298141188
// MI455X (gfx1250) — compile-verified
//
#include <hip/hip_runtime.h>
#include <hip/hip_bf16.h>

typedef float v2f __attribute__((ext_vector_type(2)));
typedef float v8f __attribute__((ext_vector_type(8)));

#define BATCH   32768
#define TSEQ    28
#define DIMX    32
#define HDIM    32
#define NSTEPS  10
#define LDS_STRIDE 34   // even (8B-aligned b64 loads), breaks lane-half bank conflicts

// ---------------------------------------------------------------- WMMA fp32
__device__ __forceinline__ v8f wmma4(v2f a, v2f b, v8f c) {
  // D = A(16x4, f32) * B(4x16, f32) + C(16x16, f32); full fp32 matrix op
  return __builtin_amdgcn_wmma_f32_16x16x4_f32(false, a, false, b, (short)0, c,
                                               false, false);
}

// ---------------------------------------------------------------- activations
// tanh(x) = 1 - 2/(e^{2x}+1); v_rcp_f32 (~1 ulp) instead of IEEE division.
__device__ __forceinline__ float tanh_fast(float x) {
  float xc = fminf(fmaxf(x, -15.f), 15.f);
  float e  = __expf(2.f * xc);                       // folds into v_mul + v_exp
  return __builtin_fmaf(-2.f, __builtin_amdgcn_rcpf(e + 1.f), 1.f);
}
__device__ __forceinline__ float sigmoid_fast(float x) {
  float xc = fminf(fmaxf(x, -30.f), 30.f);
  return __builtin_amdgcn_rcpf(1.f + __expf(-xc));
}

// ------------------------------------------------- C-layout -> A-fragments
// C layout: lane(n=L&15, half=L>>4) holds row (v+8*half), col (16*nt+n).
// A layout: lane(m=L&15, half)      holds (y[m][4kc+2half], y[m][4kc+2half+1]).
__device__ __forceinline__ void c_to_afrag(float* lds, v8f y0, v8f y1,
                                           v2f a[8], int n, int half) {
#pragma unroll
  for (int v = 0; v < 8; ++v) {
    int row = v + 8 * half;
    lds[row * LDS_STRIDE + n]      = y0[v];
    lds[row * LDS_STRIDE + 16 + n] = y1[v];
  }
  asm volatile("s_wait_dscnt 0" ::: "memory");  // cross-lane RAW through LDS
  const float* src = lds + n * LDS_STRIDE + 2 * half;  // m = n
#pragma unroll
  for (int kc = 0; kc < 8; ++kc)
    a[kc] = *(const v2f*)(src + 4 * kc);
}

// ------------------------------------------------- ODE RHS: tanh(y@W1^T+b1)@W2^T+b2
__device__ __forceinline__ void ode_rhs(const v8f y[2], v8f k[2], float* lds,
                                        const v2f W1f[2][8], const v2f W2f[2][8],
                                        const float b1v[2], const float b2v[2],
                                        int n, int half) {
  v2f a[8];
  c_to_afrag(lds, y[0], y[1], a, n, half);
  v8f c0, c1;
#pragma unroll
  for (int v = 0; v < 8; ++v) { c0[v] = b1v[0]; c1[v] = b1v[1]; }
#pragma unroll
  for (int kc = 0; kc < 8; ++kc) {
    c0 = wmma4(a[kc], W1f[0][kc], c0);
    c1 = wmma4(a[kc], W1f[1][kc], c1);
  }
  v8f t0, t1;
#pragma unroll
  for (int v = 0; v < 8; ++v) { t0[v] = tanh_fast(c0[v]); t1[v] = tanh_fast(c1[v]); }
  c_to_afrag(lds, t0, t1, a, n, half);
#pragma unroll
  for (int v = 0; v < 8; ++v) { c0[v] = b2v[0]; c1[v] = b2v[1]; }
#pragma unroll
  for (int kc = 0; kc < 8; ++kc) {
    c0 = wmma4(a[kc], W2f[0][kc], c0);
    c1 = wmma4(a[kc], W2f[1][kc], c1);
  }
  k[0] = c0; k[1] = c1;
}

// ------------------------------------------------- fixed-step RK4
__device__ __forceinline__ void rk4(v8f h[2], float span, float* lds,
                                    const v2f W1f[2][8], const v2f W2f[2][8],
                                    const float b1v[2], const float b2v[2],
                                    int n, int half) {
  const float dt = span / (float)NSTEPS;
#pragma unroll 1
  for (int s = 0; s < NSTEPS; ++s) {
    v8f k[2], acc[2], yt[2];
    ode_rhs(h, k, lds, W1f, W2f, b1v, b2v, n, half);           // k1
    acc[0] = k[0]; acc[1] = k[1];
    yt[0] = h[0] + (0.5f * dt) * k[0]; yt[1] = h[1] + (0.5f * dt) * k[1];
    ode_rhs(yt, k, lds, W1f, W2f, b1v, b2v, n, half);          // k2
    acc[0] += 2.f * k[0]; acc[1] += 2.f * k[1];
    yt[0] = h[0] + (0.5f * dt) * k[0]; yt[1] = h[1] + (0.5f * dt) * k[1];
    ode_rhs(yt, k, lds, W1f, W2f, b1v, b2v, n, half);          // k3
    acc[0] += 2.f * k[0]; acc[1] += 2.f * k[1];
    yt[0] = h[0] + dt * k[0]; yt[1] = h[1] + dt * k[1];
    ode_rhs(yt, k, lds, W1f, W2f, b1v, b2v, n, half);          // k4
    acc[0] += k[0]; acc[1] += k[1];
    h[0] += (dt / 6.f) * acc[0];
    h[1] += (dt / 6.f) * acc[1];
  }
}

// ------------------------------------------------- one 16-col gate tile of x@W^T
__device__ __forceinline__ v8f gemm_tile(const v2f a[8], const float* Wrow, float bv) {
  v8f c;
#pragma unroll
  for (int v = 0; v < 8; ++v) c[v] = bv;
#pragma unroll
  for (int kc = 0; kc < 8; ++kc) {
    v2f bw = *(const v2f*)(Wrow + 4 * kc);   // B[k][col] = W[col][k], streamed from L2
    c = wmma4(a[kc], bw, c);
  }
  return c;
}

// ------------------------------------------------- GRU cell (PyTorch gate order r,z,n)
__device__ __forceinline__ void gru_step(v8f h[2], const float* xbase,
                                         const float* Wih, const float* Whh,
                                         const float bihv[6], const float bhhv[6],
                                         float* lds, int n, int half) {
  v2f ax[8], ah[8];
  const float* xr = xbase + n * (TSEQ * DIMX) + 2 * half;  // row m = n
#pragma unroll
  for (int kc = 0; kc < 8; ++kc) ax[kc] = *(const v2f*)(xr + 4 * kc);
  c_to_afrag(lds, h[0], h[1], ah, n, half);

  const float* WihB = Wih + n * DIMX + 2 * half;
  const float* WhhB = Whh + n * HDIM + 2 * half;

  v8f r[2], z[2];
#pragma unroll
  for (int j = 0; j < 2; ++j) {            // r gate (cols 0..31)
    v8f gi = gemm_tile(ax, WihB + (0 + j) * 16 * DIMX, bihv[0 + j]);
    v8f gh = gemm_tile(ah, WhhB + (0 + j) * 16 * HDIM, bhhv[0 + j]);
#pragma unroll
    for (int v = 0; v < 8; ++v) r[j][v] = sigmoid_fast(gi[v] + gh[v]);
  }
#pragma unroll
  for (int j = 0; j < 2; ++j) {            // z gate (cols 32..63)
    v8f gi = gemm_tile(ax, WihB + (2 + j) * 16 * DIMX, bihv[2 + j]);
    v8f gh = gemm_tile(ah, WhhB + (2 + j) * 16 * HDIM, bhhv[2 + j]);
#pragma unroll
    for (int v = 0; v < 8; ++v) z[j][v] = sigmoid_fast(gi[v] + gh[v]);
  }
#pragma unroll
  for (int j = 0; j < 2; ++j) {            // n gate (cols 64..95) + state update
    v8f gi = gemm_tile(ax, WihB + (4 + j) * 16 * DIMX, bihv[4 + j]);
    v8f gh = gemm_tile(ah, WhhB + (4 + j) * 16 * HDIM, bhhv[4 + j]);
#pragma unroll
    for (int v = 0; v < 8; ++v) {
      float nn = tanh_fast(gi[v] + r[j][v] * gh[v]);
      h[j][v] = (1.f - z[j][v]) * nn + z[j][v] * h[j][v];
    }
  }
}

// ================================================================ kernel
// __launch_bounds__(256, 4): demand >=4 waves/SIMD so the allocator stays
// <=256 VGPRs -> no s_set_vgpr_msb in the hot loop, better occupancy for
// WMMA/VALU overlap across waves.
__global__ __launch_bounds__(256, 4) void node_gru_kernel(
    const float* __restrict__ inputs, const float* __restrict__ h0,
    const float* __restrict__ Wih, const float* __restrict__ Whh,
    const float* __restrict__ bih, const float* __restrict__ bhh,
    const float* __restrict__ W1, const float* __restrict__ b1,
    const float* __restrict__ W2, const float* __restrict__ b2,
    float* __restrict__ out) {
  __shared__ float lds_all[8 * 16 * LDS_STRIDE];
  const int lane = threadIdx.x & 31;
  const int wid  = threadIdx.x >> 5;
  const int n = lane & 15, half = lane >> 4;
  float* lds = lds_all + wid * (16 * LDS_STRIDE);
  const long b0 = ((long)blockIdx.x * (blockDim.x >> 5) + wid) * 16;

  // W1/W2 B-fragments resident in VGPRs (used 1160x each)
  v2f W1f[2][8], W2f[2][8];
#pragma unroll
  for (int nt = 0; nt < 2; ++nt) {
    const float* w1r = W1 + (16 * nt + n) * HDIM + 2 * half;
    const float* w2r = W2 + (16 * nt + n) * HDIM + 2 * half;
#pragma unroll
    for (int kc = 0; kc < 8; ++kc) {
      W1f[nt][kc] = *(const v2f*)(w1r + 4 * kc);
      W2f[nt][kc] = *(const v2f*)(w2r + 4 * kc);
    }
  }
  const float b1v[2] = { b1[n], b1[16 + n] };
  const float b2v[2] = { b2[n], b2[16 + n] };
  float bihv[6], bhhv[6];
#pragma unroll
  for (int g = 0; g < 6; ++g) { bihv[g] = bih[16 * g + n]; bhhv[g] = bhh[16 * g + n]; }

  // h0 tile -> C layout
  v8f h[2];
#pragma unroll
  for (int v = 0; v < 8; ++v) {
    long row = b0 + v + 8 * half;
    h[0][v] = h0[row * HDIM + n];
    h[1][v] = h0[row * HDIM + 16 + n];
  }

  rk4(h, 1.0f, lds, W1f, W2f, b1v, b2v, n, half);

  const float* xb = inputs + b0 * (TSEQ * DIMX);
#pragma unroll 1
  for (int t = 0; t < TSEQ - 1; ++t) {
    gru_step(h, xb + t * DIMX, Wih, Whh, bihv, bhhv, lds, n, half);
    rk4(h, 1.0f, lds, W1f, W2f, b1v, b2v, n, half);
  }
  gru_step(h, xb + (TSEQ - 1) * DIMX, Wih, Whh, bihv, bhhv, lds, n, half);

  // out[0] = relu(h_start)
#pragma unroll
  for (int v = 0; v < 8; ++v) {
    long row = b0 + v + 8 * half;
    out[row * HDIM + n]      = fmaxf(h[0][v], 0.f);
    out[row * HDIM + 16 + n] = fmaxf(h[1][v], 0.f);
  }

  rk4(h, 14.0f, lds, W1f, W2f, b1v, b2v, n, half);

  // out[1] = relu(h_end)
  const long off = (long)BATCH * HDIM;
#pragma unroll
  for (int v = 0; v < 8; ++v) {
    long row = b0 + v + 8 * half;
    out[off + row * HDIM + n]      = fmaxf(h[0][v], 0.f);
    out[off + row * HDIM + 16 + n] = fmaxf(h[1][v], 0.f);
  }
}

// ================================================================ launch
extern "C" void kernel_launch(void* const* d_in, const int* in_sizes, int n_in,
                              void* d_out, int out_size, void* d_ws, size_t ws_size,
                              hipStream_t stream) {
  const float* inputs = (const float*)d_in[0];
  const float* h0     = (const float*)d_in[1];
  const float* Wih    = (const float*)d_in[2];
  const float* Whh    = (const float*)d_in[3];
  const float* bih    = (const float*)d_in[4];
  const float* bhh    = (const float*)d_in[5];
  const float* W1     = (const float*)d_in[6];
  const float* b1     = (const float*)d_in[7];
  const float* W2     = (const float*)d_in[8];
  const float* b2     = (const float*)d_in[9];
  float* out = (float*)d_out;

  dim3 block(256);                       // 8 waves/block, 16 batch rows per wave
  dim3 grid(BATCH / (16 * 8));           // 256 blocks -> 2048 waves
  node_gru_kernel<<<grid, block, 0, stream>>>(inputs, h0, Wih, Whh, bih, bhh,
                                              W1, b1, W2, b2, out);
}